// DNC_9921374454418
// MI455X (gfx1250) — compile-verified
//
#include <hip/hip_runtime.h>
#include <math.h>

// ---------------- problem constants ----------------
#define TT     64
#define BB     16
#define NNM    256     // memory slots N
#define WWD    64      // word width W
#define RR     4       // read heads
#define INDIM  320     // IN + R*W
#define NHID   256
#define NNOUT  256
#define IFS    471     // interface size
#define OUTD   64
#define EPSF   1e-6f
#define NWG    16
#define NTH    256

typedef __attribute__((ext_vector_type(2))) float v2f;
typedef __attribute__((ext_vector_type(8))) float v8f;

__device__ __forceinline__ float sigmoidf(float x) { return 1.0f / (1.0f + expf(-x)); }
__device__ __forceinline__ float oneplusf(float x) {
    float sp = (x > 20.0f) ? x : log1pf(expf(x));
    return 1.0f + sp;
}

// ---------------- async global->LDS gather (CDNA5, ASYNCcnt) ----------------
// Per-lane DMA: copies one b32 from the global address in vaddr to the LDS address in vdst, GV addressing, tracked by
// ASYNCcnt; completes without touching VGPRs. The LDS offset is the low 32
// bits of the flat shared-memory pointer (ISA 10.2: LDS aperture keeps the
// group-relative offset in addr[31:0]).
template <class GA>
__device__ __forceinline__ void async_gather_to_lds(float* dst, int count, GA gaddr) {
#if defined(__HIP_DEVICE_COMPILE__)
    for (int i = threadIdx.x; i < count; i += NTH) {
        unsigned lds = (unsigned)(uintptr_t)(dst + i);
        const float* g = gaddr(i);
        asm volatile("global_load_async_to_lds_b32 %0, %1, off"
                     :: "v"(lds), "v"(g) : "memory");
    }
    asm volatile("s_wait_asynccnt 0x0" ::: "memory");
#else
    for (int i = threadIdx.x; i < count; i += NTH) dst[i] = *gaddr(i);
#endif
}

// ---------------- block-wide reductions (256 threads) ----------------
__device__ __forceinline__ float blk_sum(float v, float* red) {
    int t = threadIdx.x;
    red[t] = v; __syncthreads();
    for (int s = NTH >> 1; s > 0; s >>= 1) {
        if (t < s) red[t] += red[t + s];
        __syncthreads();
    }
    float r = red[0]; __syncthreads();
    return r;
}
__device__ __forceinline__ float blk_max(float v, float* red) {
    int t = threadIdx.x;
    red[t] = v; __syncthreads();
    for (int s = NTH >> 1; s > 0; s >>= 1) {
        if (t < s) red[t] = fmaxf(red[t], red[t + s]);
        __syncthreads();
    }
    float r = red[0]; __syncthreads();
    return r;
}
__device__ __forceinline__ void blk_sum4(float v[4], float* red4) {
    int t = threadIdx.x;
#pragma unroll
    for (int r = 0; r < 4; r++) red4[t * 4 + r] = v[r];
    __syncthreads();
    for (int s = NTH >> 1; s > 0; s >>= 1) {
        if (t < s) {
#pragma unroll
            for (int r = 0; r < 4; r++) red4[t * 4 + r] += red4[(t + s) * 4 + r];
        }
        __syncthreads();
    }
#pragma unroll
    for (int r = 0; r < 4; r++) v[r] = red4[r];
    __syncthreads();
}
__device__ __forceinline__ void blk_max4(float v[4], float* red4) {
    int t = threadIdx.x;
#pragma unroll
    for (int r = 0; r < 4; r++) red4[t * 4 + r] = v[r];
    __syncthreads();
    for (int s = NTH >> 1; s > 0; s >>= 1) {
        if (t < s) {
#pragma unroll
            for (int r = 0; r < 4; r++)
                red4[t * 4 + r] = fmaxf(red4[t * 4 + r], red4[(t + s) * 4 + r]);
        }
        __syncthreads();
    }
#pragma unroll
    for (int r = 0; r < 4; r++) v[r] = red4[r];
    __syncthreads();
}

// ---------------- grid barrier (monotone counter in d_ws) ----------------
__device__ __forceinline__ void grid_barrier(unsigned* cnt, unsigned* s_epoch) {
    __threadfence();
    __syncthreads();
    if (threadIdx.x == 0) {
        unsigned e = ++(*s_epoch);
        __hip_atomic_fetch_add(cnt, 1u, __ATOMIC_ACQ_REL, __HIP_MEMORY_SCOPE_AGENT);
        while (__hip_atomic_load(cnt, __ATOMIC_ACQUIRE, __HIP_MEMORY_SCOPE_AGENT) < e * (unsigned)NWG) {
            __builtin_amdgcn_s_sleep(4);
        }
    }
    __syncthreads();
    __threadfence();
}

// ---------------- fp32 WMMA 16x16 tile GEMM, K in steps of 4 ----------------
// A: 16(M=batch) x K ; B: K x 16 ; C/D: 16x16 fp32 accum, one wave per tile.
// VGPR layouts per CDNA5 ISA 7.12.2 (wave32):
//   A: lane&15 = M row; lanes 0-15 carry K=k,k+1; lanes 16-31 carry K=k+2,k+3
//   B: VGPR0 = rows k (lanes 0-15) / k+2 (lanes 16-31); VGPR1 = k+1 / k+3
//   D: VGPR v, lanes 0-15 -> M=v, N=lane; lanes 16-31 -> M=v+8, N=lane-16
template <class AF, class BF>
__device__ __forceinline__ v8f mm_tile(int K, AF aload, BF bload) {
    v8f acc = {};
    const int lane = threadIdx.x & 31;
    const int half = lane >> 4;
    const int nl   = lane & 15;
    for (int k = 0; k < K; k += 4) {
        const int ka = k + (half << 1);
        v2f a, b;
        a.x = aload(nl, ka);
        a.y = aload(nl, ka + 1);
        b.x = bload(ka, nl);
        b.y = bload(ka + 1, nl);
#if defined(__HIP_DEVICE_COMPILE__)
#if __has_builtin(__builtin_amdgcn_wmma_f32_16x16x4_f32)
        acc = __builtin_amdgcn_wmma_f32_16x16x4_f32(false, a, false, b, (short)0, acc,
                                                    false, false);
#else
#error "wmma_f32_16x16x4_f32 builtin not available on this device toolchain"
#endif
#else
        acc[0] += a.x * b.x;   // host pass stub (never executed)
#endif
    }
    return acc;
}

// =====================================================================
__global__ void __launch_bounds__(NTH, 1)
dnc_kernel(const float* __restrict__ inp,   // (T,B,64)
           const float* __restrict__ W1,    // (320,256)
           const float* __restrict__ b1,    // (256)
           const float* __restrict__ W2,    // (256,256)
           const float* __restrict__ b2,    // (256)
           const float* __restrict__ Wif,   // (256,471)
           const float* __restrict__ bif,   // (471)
           const float* __restrict__ Wout,  // (256,64)
           const float* __restrict__ Wmem,  // (256,64)
           float* __restrict__ out,         // (T,B,64)
           float* __restrict__ wsf,         // float scratch
           unsigned* __restrict__ bar)      // grid barrier counter
{
    // ---- scratch layout ----
    float* memG   = wsf;                       // B*N*W
    float* linkG  = memG  + BB * NNM * WWD;    // B*N*N
    float* usageG = linkG + BB * NNM * NNM;    // B*N
    float* wwG    = usageG + BB * NNM;         // B*N
    float* precG  = wwG    + BB * NNM;         // B*N
    float* rwG    = precG  + BB * NNM;         // B*N*R
    float* rvecG  = rwG    + BB * NNM * RR;    // B*256 (flat w*R+r)
    float* hG     = rvecG  + BB * RR * WWD;    // B*256
    float* nnG    = hG     + BB * NHID;        // B*256
    float* zG     = nnG    + BB * NNOUT;       // B*471

    // ---- LDS ----
    __shared__ float    s_a[BB * 512];     // staged A panel (async DMA target)
    __shared__ float    s_z[IFS];
    __shared__ float    s_er[WWD];
    __shared__ float    s_wv[WWD];
    __shared__ float    s_wwn[NNM];
    __shared__ float    s_prec[NNM];
    __shared__ float    s_rw[NNM * RR];    // old read weights
    __shared__ float    s_rwn[NNM * RR];   // new read weights
    __shared__ float    s_bwd[NNM * RR];
    __shared__ float    s_sortk[NNM];
    __shared__ unsigned s_sorti[NNM];
    __shared__ float    s_cum[NNM];
    __shared__ float    s_alloc[NNM];
    __shared__ float    s_red[NTH];
    __shared__ float    s_red4[NTH * 4];
    __shared__ unsigned s_epoch;

    const int tid = threadIdx.x;
    const int bid = blockIdx.x;
    if (tid == 0) s_epoch = 0;
    __syncthreads();

    // ---- initialize carry state (each block handles batch bid) ----
    {
        float* mB = memG  + bid * NNM * WWD;
        float* lB = linkG + bid * NNM * NNM;
        for (int i = tid; i < NNM * WWD; i += NTH) mB[i] = EPSF;
        for (int i = tid; i < NNM * NNM; i += NTH) lB[i] = 0.0f;
        for (int i = tid; i < NNM; i += NTH) {
            usageG[bid * NNM + i] = 0.0f;
            wwG[bid * NNM + i]    = 0.0f;
            precG[bid * NNM + i]  = 0.0f;
        }
        for (int i = tid; i < NNM * RR; i += NTH) rwG[bid * NNM * RR + i] = 0.0f;
        for (int i = tid; i < RR * WWD; i += NTH) rvecG[bid * RR * WWD + i] = 0.0f;
    }
    grid_barrier(bar, &s_epoch);

    for (int t = 0; t < TT; t++) {
        // ============ phase 1: h = tanh(ctrl_in @ W1 + b1)  (16 tiles) ======
        // stage ctrl_in (16 x 320) into LDS via async DMA, then WMMA from LDS
        async_gather_to_lds(s_a, BB * INDIM, [&](int i) -> const float* {
            int row = i / INDIM, k = i - row * INDIM;
            return (k < 64) ? (inp + t * BB * 64 + row * 64 + k)
                            : (rvecG + row * 256 + (k - 64));
        });
        __syncthreads();
        if ((tid >> 5) == 0) {
            const int col0 = bid * 16;
            __builtin_prefetch(W1 + col0, 0, 1);
            v8f c = mm_tile(INDIM,
                [&](int row, int k) -> float { return s_a[row * INDIM + k]; },
                [&](int k, int cc) -> float { return W1[k * NHID + col0 + cc]; });
            union { v8f v; float f[8]; } u; u.v = c;
            const int lane = tid & 31, half = lane >> 4, nl = lane & 15;
#pragma unroll
            for (int v = 0; v < 8; v++) {
                int row = half * 8 + v, col = col0 + nl;
                hG[row * NHID + col] = tanhf(u.f[v] + b1[col]);
            }
        }
        grid_barrier(bar, &s_epoch);

        // ============ phase 2: nn_out = tanh(h @ W2 + b2)  (16 tiles) =======
        async_gather_to_lds(s_a, BB * NHID,
                            [&](int i) -> const float* { return hG + i; });
        __syncthreads();
        if ((tid >> 5) == 0) {
            const int col0 = bid * 16;
            v8f c = mm_tile(NHID,
                [&](int row, int k) -> float { return s_a[row * NHID + k]; },
                [&](int k, int cc) -> float { return W2[k * NNOUT + col0 + cc]; });
            union { v8f v; float f[8]; } u; u.v = c;
            const int lane = tid & 31, half = lane >> 4, nl = lane & 15;
#pragma unroll
            for (int v = 0; v < 8; v++) {
                int row = half * 8 + v, col = col0 + nl;
                nnG[row * NNOUT + col] = tanhf(u.f[v] + b2[col]);
            }
        }
        grid_barrier(bar, &s_epoch);

        // ============ phase 3: z = nn_out @ W_if + b_if  (30 tiles) =========
        async_gather_to_lds(s_a, BB * NNOUT,
                            [&](int i) -> const float* { return nnG + i; });
        __syncthreads();
        {
            const int wv = tid >> 5;
            const int tile = (wv == 0) ? bid : ((wv == 1) ? bid + 16 : 999);
            if (tile < 30) {
                const int col0 = tile * 16;
                v8f c = mm_tile(NNOUT,
                    [&](int row, int k) -> float { return s_a[row * NNOUT + k]; },
                    [&](int k, int cc) -> float {
                        int col = col0 + cc;
                        return (col < IFS) ? Wif[k * IFS + col] : 0.0f;
                    });
                union { v8f v; float f[8]; } u; u.v = c;
                const int lane = tid & 31, half = lane >> 4, nl = lane & 15;
#pragma unroll
                for (int v = 0; v < 8; v++) {
                    int row = half * 8 + v, col = col0 + nl;
                    if (col < IFS) zG[row * IFS + col] = u.f[v] + bif[col];
                }
            }
        }
        grid_barrier(bar, &s_epoch);

        // ============ phase 4: per-batch memory machinery ====================
        {
            const int b = bid;
            const int n = tid;                 // memory slot index
            float* memB  = memG  + b * NNM * WWD;
            float* linkB = linkG + b * NNM * NNM;

            // interface row -> LDS via async DMA
            async_gather_to_lds(s_z, IFS,
                                [&](int i) -> const float* { return zG + b * IFS + i; });
            __syncthreads();
            if (n < WWD) { s_er[n] = sigmoidf(s_z[325 + n]); s_wv[n] = s_z[389 + n]; }

            float u_old  = usageG[b * NNM + n];
            float ww_old = wwG[b * NNM + n];
            s_prec[n] = precG[b * NNM + n];
            float rwo[4];
#pragma unroll
            for (int r = 0; r < 4; r++) {
                rwo[r] = rwG[b * NNM * RR + n * RR + r];
                s_rw[n * RR + r] = rwo[r];
            }
            __syncthreads();

            // gates / strengths
            float fg[4];
#pragma unroll
            for (int r = 0; r < 4; r++) fg[r] = sigmoidf(s_z[453 + r]);
            const float ag   = sigmoidf(s_z[457]);
            const float wg   = sigmoidf(s_z[458]);
            const float wstr = oneplusf(s_z[324]);
            float rstr[4];
#pragma unroll
            for (int r = 0; r < 4; r++) rstr[r] = oneplusf(s_z[256 + r]);

            // usage update
            float ret = 1.0f;
#pragma unroll
            for (int r = 0; r < 4; r++) ret *= (1.0f - fg[r] * rwo[r]);
            float usg = (u_old + ww_old - u_old * ww_old) * ret;
            s_sortk[n] = usg; s_sorti[n] = (unsigned)n;

            // stable ascending bitonic sort by (usage, index)
            for (unsigned ksz = 2; ksz <= (unsigned)NNM; ksz <<= 1) {
                for (unsigned j = ksz >> 1; j > 0; j >>= 1) {
                    __syncthreads();
                    unsigned i = (unsigned)n, p = i ^ j;
                    if (p > i) {
                        float    a  = s_sortk[i], c2 = s_sortk[p];
                        unsigned ia = s_sorti[i], ib = s_sorti[p];
                        bool up   = ((i & ksz) == 0);
                        bool agtb = (a > c2) || (a == c2 && ia > ib);
                        if (up == agtb) {
                            s_sortk[i] = c2; s_sortk[p] = a;
                            s_sorti[i] = ib; s_sorti[p] = ia;
                        }
                    }
                }
            }
            __syncthreads();

            // inclusive product scan -> exclusive cumprod -> allocation
            s_cum[n] = s_sortk[n]; __syncthreads();
            for (int off = 1; off < NNM; off <<= 1) {
                float v = (n >= off) ? s_cum[n - off] : 1.0f;
                __syncthreads();
                s_cum[n] *= v;
                __syncthreads();
            }
            float excl = (n == 0) ? 1.0f : s_cum[n - 1];
            s_alloc[s_sorti[n]] = (1.0f - s_sortk[n]) * excl;
            __syncthreads();

            // write content weighting on OLD memory
            float dot = 0.0f, nm2 = 0.0f;
            {
                const float* mrow = memB + n * WWD;
                for (int j = 0; j < WWD; j++) {
                    float m = mrow[j];
                    dot += m * s_z[260 + j];
                    nm2 += m * m;
                }
            }
            float wk2 = (n < WWD) ? s_z[260 + n] * s_z[260 + n] : 0.0f;
            float wkn = sqrtf(blk_sum(wk2, s_red)) + EPSF;
            float simw = dot / ((sqrtf(nm2) + EPSF) * wkn) * wstr;
            float mxw  = blk_max(simw, s_red);
            float exw  = expf(simw - mxw);
            float smw  = blk_sum(exw, s_red);
            float wc   = exw / smw;

            float wwn = wg * (ag * s_alloc[n] + (1.0f - ag) * wc);
            s_wwn[n] = wwn;
            __syncthreads();

            // read key norms
            float rk2[4];
#pragma unroll
            for (int r = 0; r < 4; r++)
                rk2[r] = (n < WWD) ? s_z[n * RR + r] * s_z[n * RR + r] : 0.0f;
            blk_sum4(rk2, s_red4);
            float rkn[4];
#pragma unroll
            for (int r = 0; r < 4; r++) rkn[r] = sqrtf(rk2[r]) + EPSF;

            // memory write (erase/add) fused with read-content dots on NEW mem
            float dr[4] = {0, 0, 0, 0};
            float nn2 = 0.0f;
            {
                float* mrw = memB + n * WWD;
                for (int j = 0; j < WWD; j++) {
                    float m = mrw[j];
                    m = m * (1.0f - wwn * s_er[j]) + wwn * s_wv[j];
                    mrw[j] = m;
                    nn2 += m * m;
#pragma unroll
                    for (int r = 0; r < 4; r++) dr[r] += m * s_z[j * RR + r];
                }
            }
            float mn = sqrtf(nn2) + EPSF;
            float simr[4], mx4[4];
#pragma unroll
            for (int r = 0; r < 4; r++) { simr[r] = dr[r] / (mn * rkn[r]) * rstr[r]; mx4[r] = simr[r]; }
            blk_max4(mx4, s_red4);
            float er4[4], s4[4];
#pragma unroll
            for (int r = 0; r < 4; r++) { er4[r] = expf(simr[r] - mx4[r]); s4[r] = er4[r]; }
            blk_sum4(s4, s_red4);
            float rc[4];
#pragma unroll
            for (int r = 0; r < 4; r++) rc[r] = er4[r] / s4[r];
            __syncthreads();

            // link pass 1: column-coalesced update + backward einsum
            {
                float precm = s_prec[n];
                float wwm   = s_wwn[n];
                float bw[4] = {0, 0, 0, 0};
                for (int row = 0; row < NNM; row++) {
                    float l   = linkB[row * NNM + n];
                    float wwr = s_wwn[row];
                    float ln  = (1.0f - wwr - wwm) * l + wwr * precm;
                    if (row == n) ln = 0.0f;
                    linkB[row * NNM + n] = ln;
#pragma unroll
                    for (int r = 0; r < 4; r++) bw[r] += ln * s_rw[row * RR + r];
                }
#pragma unroll
                for (int r = 0; r < 4; r++) s_bwd[n * RR + r] = bw[r];
            }
            __threadfence();
            __syncthreads();

            // link pass 2: row-wise forward einsum on NEW link
            float fw[4] = {0, 0, 0, 0};
            {
                const float* lrow = linkB + n * NNM;
                for (int m2 = 0; m2 < NNM; m2++) {
                    float l = lrow[m2];
#pragma unroll
                    for (int r = 0; r < 4; r++) fw[r] += l * s_rw[m2 * RR + r];
                }
            }

            // read modes: softmax over 3 per head
            float rm0[4], rm1[4], rm2[4];
#pragma unroll
            for (int r = 0; r < 4; r++) {
                float a0 = s_z[459 + r], a1 = s_z[463 + r], a2 = s_z[467 + r];
                float mm = fmaxf(a0, fmaxf(a1, a2));
                float e0 = expf(a0 - mm), e1 = expf(a1 - mm), e2 = expf(a2 - mm);
                float ss = e0 + e1 + e2;
                rm0[r] = e0 / ss; rm1[r] = e1 / ss; rm2[r] = e2 / ss;
            }

            // new read weights
#pragma unroll
            for (int r = 0; r < 4; r++) {
                float v = s_bwd[n * RR + r] * rm0[r] + rc[r] * rm1[r] + fw[r] * rm2[r];
                s_rwn[n * RR + r] = v;
                rwG[b * NNM * RR + n * RR + r] = v;
            }

            // precedence + state writeback
            float wsum = blk_sum(wwn, s_red);
            precG[b * NNM + n]  = (1.0f - wsum) * s_prec[n] + wwn;
            usageG[b * NNM + n] = usg;
            wwG[b * NNM + n]    = wwn;
            __threadfence();
            __syncthreads();

            // read vectors: rvec[w][r] = sum_n mem[n][w]*rw_new[n][r]
            {
                int w = n >> 2, r2 = n & 3;
                float acc = 0.0f;
                for (int n2 = 0; n2 < NNM; n2++)
                    acc += memB[n2 * WWD + w] * s_rwn[n2 * RR + r2];
                rvecG[b * 256 + n] = acc;   // flat index w*R+r == n
            }
        }
        grid_barrier(bar, &s_epoch);

        // ============ phase 5: out = nn_out@W_out + rvec@W_mem_out (4 tiles)
        if (bid < 4) {
            async_gather_to_lds(s_a, BB * 512, [&](int i) -> const float* {
                int row = i >> 9, k = i & 511;
                return (k < 256) ? (nnG + row * 256 + k)
                                 : (rvecG + row * 256 + (k - 256));
            });
            __syncthreads();
            if ((tid >> 5) == 0) {
                const int col0 = bid * 16;
                v8f c = mm_tile(512,
                    [&](int row, int k) -> float { return s_a[row * 512 + k]; },
                    [&](int k, int cc) -> float {
                        int col = col0 + cc;
                        return (k < 256) ? Wout[k * OUTD + col]
                                         : Wmem[(k - 256) * OUTD + col];
                    });
                union { v8f v; float f[8]; } u; u.v = c;
                const int lane = tid & 31, half = lane >> 4, nl = lane & 15;
#pragma unroll
                for (int v = 0; v < 8; v++) {
                    int row = half * 8 + v, col = col0 + nl;
                    out[t * BB * OUTD + row * OUTD + col] = u.f[v];
                }
            }
        }
        // no trailing barrier needed: next phase-1 reads only inp/rvecG,
        // and rvecG is not rewritten until after 3 more grid barriers.
    }
}

// =====================================================================
extern "C" void kernel_launch(void* const* d_in, const int* in_sizes, int n_in,
                              void* d_out, int out_size, void* d_ws, size_t ws_size,
                              hipStream_t stream) {
    (void)in_sizes; (void)n_in; (void)out_size; (void)ws_size;
    const float* inp  = (const float*)d_in[0];
    const float* W1   = (const float*)d_in[1];
    const float* b1   = (const float*)d_in[2];
    const float* W2   = (const float*)d_in[3];
    const float* b2   = (const float*)d_in[4];
    const float* Wif  = (const float*)d_in[5];
    const float* bif  = (const float*)d_in[6];
    const float* Wout = (const float*)d_in[7];
    const float* Wmem = (const float*)d_in[8];
    float*    out = (float*)d_out;
    unsigned* bar = (unsigned*)d_ws;
    float*    wsf = (float*)((char*)d_ws + 256);

    // reset grid-barrier counter each call (graph-capture legal)
    (void)hipMemsetAsync(d_ws, 0, 256, stream);

    dnc_kernel<<<NWG, NTH, 0, stream>>>(inp, W1, b1, W2, b2, Wif, bif, Wout, Wmem,
                                        out, wsf, bar);
}